// GaussianPointMatcher_40767829574324
// MI455X (gfx1250) — compile-verified
//
#include <hip/hip_runtime.h>
#include <hip/hip_bf16.h>
#include <float.h>

// ---------------------------------------------------------------------------
// GaussianPointMatcher for MI455X (gfx1250, wave32, WMMA + async-to-LDS).
//
// Pipeline:
//   1. k_init / k_minmax : per-axis min/max of pointcloud (ordered-uint atomics)
//   2. k_morton          : 30-bit Morton codes, key = (code<<32)|index (stable)
//   3. k_sort            : single-block bitonic sort of 64-bit keys (L2 resident)
//   4. k_gather          : sorted_points -> d_out[0 .. 3P)
//   5. k_gauss           : quat -> R, A = R diag(1/s^2) R^T, b = A mu, c = b.mu
//   6. k_median          : median(radius_x) via O(G^2) rank count -> voxel
//   7. k_bounds          : gmin/gmax voxel box + oversized flag per gaussian
//   8. k_pack            : per-gaussian-tile blob [coef 1KB | bounds 512B | pad]
//   9. k_feat            : feat[p][16] = [p⊗p(9), p(3), 1, 0,0,0]; point voxel
//  10. k_match           : q[p,g] = feat·coef via V_WMMA_F32_16X16X4_F32 (K=16
//                          as 4 chained K=4 steps).  Gaussian tiles streamed
//                          into LDS with double-buffered
//                          GLOBAL_LOAD_ASYNC_TO_LDS_B64 + s_wait_asynccnt.
//                          Log-domain argmax (track min q), one expf per point.
// ---------------------------------------------------------------------------

typedef float v2f __attribute__((ext_vector_type(2)));
typedef float v8f __attribute__((ext_vector_type(8)));

#define SQRT_CONF_K 2.7954834216f   // sqrt(7.814727903)
#define VOXEL_FACTOR 3.0f
#define OVERSIZED_VOXELS 64

// ---- float <-> order-preserving uint ----
__device__ __forceinline__ unsigned encF(float f) {
    unsigned u = __float_as_uint(f);
    return (u & 0x80000000u) ? ~u : (u | 0x80000000u);
}
__device__ __forceinline__ float decF(unsigned s) {
    unsigned u = (s & 0x80000000u) ? (s & 0x7FFFFFFFu) : ~s;
    return __uint_as_float(u);
}

__device__ __forceinline__ unsigned spread_bits(unsigned v) {
    v = (v | (v << 16)) & 0x030000FFu;
    v = (v | (v << 8))  & 0x0300F00Fu;
    v = (v | (v << 4))  & 0x030C30C3u;
    v = (v | (v << 2))  & 0x09249249u;
    return v;
}

// CDNA5 async copy: 8 bytes per lane, global -> LDS, tracked by ASYNCcnt.
// GVS addressing: mem = SADDR(64) + VADDR(i32) ; LDS dest = VDST lane value.
__device__ __forceinline__ void async_copy_b64(unsigned lds_addr, const void* sbase,
                                               unsigned voff) {
    asm volatile("global_load_async_to_lds_b64 %0, %1, %2"
                 :: "v"(lds_addr), "v"(voff), "s"(sbase)
                 : "memory");
}

// ---------------------------------------------------------------------------
__global__ void gpm_init(unsigned* mm) {
    int i = threadIdx.x;
    if (i < 3) mm[i] = 0xFFFFFFFFu;       // encoded mins
    else if (i < 6) mm[i] = 0u;           // encoded maxs
}

__global__ __launch_bounds__(256) void gpm_minmax(const float* __restrict__ pts,
                                                  unsigned* mm, int P) {
    int p = blockIdx.x * 256 + threadIdx.x;
    if (p >= P) return;
    unsigned en[3], ex[3];
#pragma unroll
    for (int i = 0; i < 3; ++i) { unsigned e = encF(pts[p * 3 + i]); en[i] = e; ex[i] = e; }
#pragma unroll
    for (int o = 16; o; o >>= 1) {
#pragma unroll
        for (int i = 0; i < 3; ++i) {
            unsigned a = __shfl_xor(en[i], o); if (a < en[i]) en[i] = a;
            unsigned b = __shfl_xor(ex[i], o); if (b > ex[i]) ex[i] = b;
        }
    }
    if ((threadIdx.x & 31) == 0) {
#pragma unroll
        for (int i = 0; i < 3; ++i) { atomicMin(&mm[i], en[i]); atomicMax(&mm[3 + i], ex[i]); }
    }
}

__global__ __launch_bounds__(256) void gpm_morton(const float* __restrict__ pts,
                                                  const unsigned* __restrict__ mm,
                                                  unsigned long long* __restrict__ keys, int P) {
    int p = blockIdx.x * 256 + threadIdx.x;
    if (p >= P) return;
    unsigned u[3];
#pragma unroll
    for (int i = 0; i < 3; ++i) {
        float mn = decF(mm[i]), mx = decF(mm[3 + i]);
        float t = (pts[p * 3 + i] - mn) / (mx - mn + 1e-9f) * 1023.0f;
        t = fminf(fmaxf(t, 0.0f), 1023.0f);
        u[i] = (unsigned)t;
    }
    unsigned code = spread_bits(u[0]) | (spread_bits(u[1]) << 1) | (spread_bits(u[2]) << 2);
    keys[p] = ((unsigned long long)code << 32) | (unsigned)p;   // stable argsort
}

// single-block bitonic sort over n (power of two) 64-bit keys in global memory
__global__ __launch_bounds__(1024) void gpm_sort(unsigned long long* __restrict__ keys, int n) {
    const unsigned tid = threadIdx.x;
    for (unsigned k = 2; k <= (unsigned)n; k <<= 1) {
        for (unsigned j = k >> 1; j > 0; j >>= 1) {
            for (unsigned i = tid; i < (unsigned)n; i += 1024u) {
                unsigned ixj = i ^ j;
                if (ixj > i) {
                    unsigned long long a = keys[i];
                    unsigned long long b = keys[ixj];
                    bool up = ((i & k) == 0);
                    if ((a > b) == up) { keys[i] = b; keys[ixj] = a; }
                }
            }
            __syncthreads();
        }
    }
}

__global__ __launch_bounds__(256) void gpm_gather(const unsigned long long* __restrict__ keys,
                                                  const float* __restrict__ pts,
                                                  float* __restrict__ outPts, int P) {
    int p = blockIdx.x * 256 + threadIdx.x;
    if (p >= P) return;
    unsigned idx = (unsigned)keys[p];
    outPts[p * 3 + 0] = pts[(size_t)idx * 3 + 0];
    outPts[p * 3 + 1] = pts[(size_t)idx * 3 + 1];
    outPts[p * 3 + 2] = pts[(size_t)idx * 3 + 2];
}

// per-gaussian: A = Sigma^-1, coef row [A(9), -2b(3), c, 0,0,0], radius(3)
__global__ __launch_bounds__(256) void gpm_gauss(const float* __restrict__ pos,
                                                 const float* __restrict__ scl,
                                                 const float* __restrict__ quat,
                                                 float* __restrict__ coef,
                                                 float4* __restrict__ rad, int G) {
    int g = blockIdx.x * 256 + threadIdx.x;
    if (g >= G) return;
    float w = quat[g * 4 + 0], x = quat[g * 4 + 1], y = quat[g * 4 + 2], z = quat[g * 4 + 3];
    float inv = rsqrtf(w * w + x * x + y * y + z * z);
    w *= inv; x *= inv; y *= inv; z *= inv;
    float R00 = 1.f - 2.f * (y * y + z * z), R01 = 2.f * (x * y - w * z), R02 = 2.f * (x * z + w * y);
    float R10 = 2.f * (x * y + w * z), R11 = 1.f - 2.f * (x * x + z * z), R12 = 2.f * (y * z - w * x);
    float R20 = 2.f * (x * z - w * y), R21 = 2.f * (y * z + w * x), R22 = 1.f - 2.f * (x * x + y * y);
    float s0 = scl[g * 3 + 0], s1 = scl[g * 3 + 1], s2 = scl[g * 3 + 2];
    float v0 = s0 * s0, v1 = s1 * s1, v2 = s2 * s2;
    float i0 = 1.f / v0, i1 = 1.f / v1, i2 = 1.f / v2;
    float A00 = R00 * R00 * i0 + R01 * R01 * i1 + R02 * R02 * i2;
    float A01 = R00 * R10 * i0 + R01 * R11 * i1 + R02 * R12 * i2;
    float A02 = R00 * R20 * i0 + R01 * R21 * i1 + R02 * R22 * i2;
    float A11 = R10 * R10 * i0 + R11 * R11 * i1 + R12 * R12 * i2;
    float A12 = R10 * R20 * i0 + R11 * R21 * i1 + R12 * R22 * i2;
    float A22 = R20 * R20 * i0 + R21 * R21 * i1 + R22 * R22 * i2;
    float px = pos[g * 3 + 0], py = pos[g * 3 + 1], pz = pos[g * 3 + 2];
    float b0 = A00 * px + A01 * py + A02 * pz;
    float b1 = A01 * px + A11 * py + A12 * pz;
    float b2 = A02 * px + A12 * py + A22 * pz;
    float c = b0 * px + b1 * py + b2 * pz;
    float4* cr = (float4*)(coef + (size_t)g * 16);
    cr[0] = make_float4(A00, A01, A02, A01);
    cr[1] = make_float4(A11, A12, A02, A12);
    cr[2] = make_float4(A22, -2.f * b0, -2.f * b1, -2.f * b2);
    cr[3] = make_float4(c, 0.f, 0.f, 0.f);
    float d0 = R00 * R00 * v0 + R01 * R01 * v1 + R02 * R02 * v2;
    float d1 = R10 * R10 * v0 + R11 * R11 * v1 + R12 * R12 * v2;
    float d2 = R20 * R20 * v0 + R21 * R21 * v1 + R22 * R22 * v2;
    rad[g] = make_float4(SQRT_CONF_K * sqrtf(d0), SQRT_CONF_K * sqrtf(d1),
                         SQRT_CONF_K * sqrtf(d2), 0.f);
}

// median of radius_x via rank counting (ties write identical values -> race-free)
__global__ __launch_bounds__(256) void gpm_median(const float4* __restrict__ rad,
                                                  float* __restrict__ med, int G) {
    int g = blockIdx.x * 256 + threadIdx.x;
    if (g >= G) return;
    float v = rad[g].x;
    int less = 0, eq = 0;
    for (int j = 0; j < G; ++j) {
        float w = rad[j].x;
        less += (w < v);
        eq += (w == v);
    }
    int k0 = (G - 1) >> 1, k1 = G >> 1;
    if (less <= k0 && k0 < less + eq) med[0] = v;
    if (less <= k1 && k1 < less + eq) med[1] = v;
}

__global__ __launch_bounds__(256) void gpm_bounds(const float* __restrict__ pos,
                                                  const float4* __restrict__ rad,
                                                  const float* __restrict__ med,
                                                  int4* __restrict__ bounds, int G) {
    int g = blockIdx.x * 256 + threadIdx.x;
    if (g >= G) return;
    float voxel = 0.5f * (med[0] + med[1]) * VOXEL_FACTOR;
    float4 r = rad[g];
    float px = pos[g * 3 + 0], py = pos[g * 3 + 1], pz = pos[g * 3 + 2];
    int mnx = (int)floorf((px - r.x) / voxel), mxx = (int)floorf((px + r.x) / voxel);
    int mny = (int)floorf((py - r.y) / voxel), mxy = (int)floorf((py + r.y) / voxel);
    int mnz = (int)floorf((pz - r.z) / voxel), mxz = (int)floorf((pz + r.z) / voxel);
    int ex = mxx - mnx + 1, ey = mxy - mny + 1, ez = mxz - mnz + 1;
    int oversized = (ex * ey * ez) > OVERSIZED_VOXELS;
    bounds[g * 2 + 0] = make_int4(mnx, mny, mnz, oversized);
    bounds[g * 2 + 1] = make_int4(mxx, mxy, mxz, 0);
}

// Per-gaussian-tile streaming blob: 512 dwords = [coef 256 | bounds 128 | pad 128]
__global__ __launch_bounds__(256) void gpm_pack(const float* __restrict__ coef,
                                                const int4* __restrict__ bounds,
                                                unsigned* __restrict__ blob, int G) {
    int o = blockIdx.x * 256 + threadIdx.x;
    int total = (G >> 4) * 512;
    if (o >= total) return;
    int gt = o >> 9, r = o & 511;
    unsigned v = 0;
    if (r < 256)      v = __float_as_uint(coef[(size_t)gt * 256 + r]);
    else if (r < 384) v = (unsigned)((const int*)bounds)[gt * 128 + (r - 256)];
    blob[o] = v;
}

// feat[p][16] = [xx,xy,xz, yx,yy,yz, zx,zy,zz, x,y,z, 1, 0,0,0]; voxel of point
__global__ __launch_bounds__(256) void gpm_feat(const float* __restrict__ sp,
                                                const float* __restrict__ med,
                                                float* __restrict__ feat,
                                                int4* __restrict__ pg, int P) {
    int p = blockIdx.x * 256 + threadIdx.x;
    if (p >= P) return;
    float voxel = 0.5f * (med[0] + med[1]) * VOXEL_FACTOR;
    float x = sp[p * 3 + 0], y = sp[p * 3 + 1], z = sp[p * 3 + 2];
    pg[p] = make_int4((int)floorf(x / voxel), (int)floorf(y / voxel), (int)floorf(z / voxel), 0);
    float4* f = (float4*)(feat + (size_t)p * 16);
    f[0] = make_float4(x * x, x * y, x * z, x * y);
    f[1] = make_float4(y * y, y * z, x * z, y * z);
    f[2] = make_float4(z * z, x, y, z);
    f[3] = make_float4(1.f, 0.f, 0.f, 0.f);
}

// ---------------------------------------------------------------------------
// Main WMMA kernel.  Wave tile: M=16 gaussians x N=16 points, K=16 features
// as 4x V_WMMA_F32_16X16X4_F32.  Gaussian tiles (2 KB blobs) are streamed
// into a double-buffered LDS staging area with GLOBAL_LOAD_ASYNC_TO_LDS_B64
// (one b64 per lane per tile -> uniform ASYNCcnt per wave; async loads retire
// in order, so s_wait_asynccnt 0x1 completes the current tile while the next
// tile's load remains in flight).
// ---------------------------------------------------------------------------
__global__ __launch_bounds__(256) void gpm_match(const float* __restrict__ feat,
                                                 const int4* __restrict__ pg,
                                                 const char* __restrict__ blob,
                                                 float* __restrict__ outBest,
                                                 int* __restrict__ outRel, int G) {
    const int tid = threadIdx.x;
    const int wave = tid >> 5, lane = tid & 31;
    const int half = lane >> 4, l16 = lane & 15;
    const int pbase0 = blockIdx.x * 128;

    __shared__ __align__(16) float featL[128 * 16];                 // 8 KB
    __shared__ __align__(16) unsigned long long tileBuf[2][256];    // 2 x 2 KB

    // stage 128 point-feature rows: 8 floats per thread
    {
        int row = tid >> 1, seg = (tid & 1) * 8;
        const float4* src = (const float4*)(feat + (size_t)(pbase0 + row) * 16 + seg);
        float4* dst = (float4*)(featL + row * 16 + seg);
        dst[0] = src[0];
        dst[1] = src[1];
    }

    // per-thread LDS destinations for the async tile stream
    unsigned ldsAddr[2];
    ldsAddr[0] = (unsigned)(uintptr_t)&tileBuf[0][tid];   // low 32 bits serve as the LDS address
    ldsAddr[1] = (unsigned)(uintptr_t)&tileBuf[1][tid];
    const unsigned voff = (unsigned)tid * 8u;

    const int tiles = G >> 4;
    // prologue: tile 0 -> buffer 0
    async_copy_b64(ldsAddr[0], blob, voff);

    __syncthreads();   // featL visible to all waves

    const int pl = wave * 16 + l16;             // block-local point index
    v2f bfrag[4];
#pragma unroll
    for (int t = 0; t < 4; ++t)
        bfrag[t] = *(const v2f*)(featL + pl * 16 + 4 * t + 2 * half);
    const int4 mypg = pg[pbase0 + pl];

    float qbest[8];
    int gbest[8];
#pragma unroll
    for (int r = 0; r < 8; ++r) { qbest[r] = FLT_MAX; gbest[r] = -1; }

    for (int gt = 0; gt < tiles; ++gt) {
        const int cur = gt & 1;
        if (gt + 1 < tiles) {
            // issue next tile into the other buffer (safe: trailing barrier of
            // iteration gt-1 guarantees no wave still reads it)
            async_copy_b64(ldsAddr[cur ^ 1], blob + (size_t)(gt + 1) * 2048, voff);
            asm volatile("s_wait_asynccnt 0x1" ::: "memory");  // retire tile gt
        } else {
            asm volatile("s_wait_asynccnt 0x0" ::: "memory");
        }
        __syncthreads();   // tile gt resident in LDS for every wave

        const float* coefL = (const float*)&tileBuf[cur][0];
        const int4*  boundL = (const int4*)((const char*)&tileBuf[cur][0] + 1024);

        v8f acc = {0.f, 0.f, 0.f, 0.f, 0.f, 0.f, 0.f, 0.f};
#pragma unroll
        for (int t = 0; t < 4; ++t) {
            v2f a = *(const v2f*)(coefL + l16 * 16 + 4 * t + 2 * half);
            acc = __builtin_amdgcn_wmma_f32_16x16x4_f32(
                false, a, false, bfrag[t], (short)0, acc, false, false);
        }

#pragma unroll
        for (int r = 0; r < 8; ++r) {
            int gl = r + 8 * half;
            int4 b0 = boundL[gl * 2 + 0];
            int4 b1 = boundL[gl * 2 + 1];
            bool inb = (mypg.x >= b0.x) & (mypg.y >= b0.y) & (mypg.z >= b0.z) &
                       (mypg.x <= b1.x) & (mypg.y <= b1.y) & (mypg.z <= b1.z);
            bool elig = (b0.w != 0) | inb;
            float q = acc[r];
            if (elig && (q < qbest[r])) { qbest[r] = q; gbest[r] = gt * 16 + gl; }
        }
        __syncthreads();   // all waves done with buffer cur before it is refilled
    }

    // reduce 8 slots (prefer smaller q, then smaller g — matches first-argmax)
    float q = qbest[0];
    int g = gbest[0];
#pragma unroll
    for (int r = 1; r < 8; ++r) {
        if (qbest[r] < q || (qbest[r] == q && (unsigned)gbest[r] < (unsigned)g)) {
            q = qbest[r]; g = gbest[r];
        }
    }
    // cross-half: lanes k and k+16 hold the same point
    float q2 = __shfl_xor(q, 16);
    int g2 = __shfl_xor(g, 16);
    if (q2 < q || (q2 == q && (unsigned)g2 < (unsigned)g)) { q = q2; g = g2; }

    if (lane < 16) {
        int p = pbase0 + wave * 16 + lane;
        float bv = (g >= 0) ? expf(-0.5f * q) : 0.0f;
        outBest[p] = bv;
        outRel[p] = (bv > 0.0f) ? g : -1;
    }
}

// ---------------------------------------------------------------------------
extern "C" void kernel_launch(void* const* d_in, const int* in_sizes, int n_in,
                              void* d_out, int out_size, void* d_ws, size_t ws_size,
                              hipStream_t stream) {
    const float* positions = (const float*)d_in[0];
    const float* scales    = (const float*)d_in[1];
    const float* quats     = (const float*)d_in[2];
    const float* pts       = (const float*)d_in[3];
    const int G = in_sizes[0] / 3;
    const int P = in_sizes[3] / 3;
    const int tiles = G / 16;

    char* ws = (char*)d_ws;
    size_t off = 0;
    auto take = [&](size_t bytes) -> char* {
        char* p = ws + off;
        off += (bytes + 255) & ~(size_t)255;
        return p;
    };
    unsigned* mm                 = (unsigned*)take(64);
    float* med                   = (float*)take(64);
    unsigned long long* keys     = (unsigned long long*)take(sizeof(unsigned long long) * (size_t)P);
    float* coef                  = (float*)take(sizeof(float) * 16 * (size_t)G);
    float4* rad                  = (float4*)take(sizeof(float4) * (size_t)G);
    int4* bounds                 = (int4*)take(sizeof(int4) * 2 * (size_t)G);
    unsigned* blob               = (unsigned*)take(2048u * (size_t)tiles);
    int4* pg                     = (int4*)take(sizeof(int4) * (size_t)P);
    float* feat                  = (float*)take(sizeof(float) * 16 * (size_t)P);

    float* outPts  = (float*)d_out;
    int*   outRel  = (int*)d_out + (size_t)3 * P;
    float* outBest = (float*)d_out + (size_t)4 * P;

    const int pb = (P + 255) / 256;
    const int gb = (G + 255) / 256;
    const int bb = (tiles * 512 + 255) / 256;

    gpm_init<<<1, 64, 0, stream>>>(mm);
    gpm_minmax<<<pb, 256, 0, stream>>>(pts, mm, P);
    gpm_morton<<<pb, 256, 0, stream>>>(pts, mm, keys, P);
    gpm_sort<<<1, 1024, 0, stream>>>(keys, P);
    gpm_gather<<<pb, 256, 0, stream>>>(keys, pts, outPts, P);
    gpm_gauss<<<gb, 256, 0, stream>>>(positions, scales, quats, coef, rad, G);
    gpm_median<<<gb, 256, 0, stream>>>(rad, med, G);
    gpm_bounds<<<gb, 256, 0, stream>>>(positions, rad, med, bounds, G);
    gpm_pack<<<bb, 256, 0, stream>>>(coef, bounds, blob, G);
    gpm_feat<<<pb, 256, 0, stream>>>(outPts, med, feat, pg, P);
    gpm_match<<<P / 128, 256, 0, stream>>>(feat, pg, (const char*)blob, outBest, outRel, G);
}